// SelectiveScanTorch_6828998001039
// MI455X (gfx1250) — compile-verified
//
#include <hip/hip_runtime.h>
#include <stdint.h>

namespace {
constexpr int BB = 2;                  // batch
constexpr int DD = 1024;               // d-model
constexpr int LL = 1024;               // seq len
constexpr int NN = 16;                 // state dim
constexpr int WPB = 8;                 // waves per block
constexpr int THREADS = WPB * 32;      // 256 threads = 8 wave32
constexpr int ILP = 8;                 // l-values per lane per stage
constexpr int CHUNK = 32 * ILP;        // 256 l-values per pipeline stage
constexpr int NCHUNK = LL / CHUNK;     // 4
constexpr int V4_PER_MAT = NN * CHUNK / 4;            // 1024 float4 per B (or C) panel
constexpr int V4_PER_THREAD = V4_PER_MAT / THREADS;   // 4
}

__device__ __forceinline__ float softplus_f(float x) {
  return (x > 20.0f) ? x : __logf(1.0f + __expf(x));
}

__device__ __forceinline__ float sigmoid_f(float x) {
  return 1.0f / (1.0f + __expf(-x));
}

__global__ __launch_bounds__(THREADS)
void selscan_kernel(const float* __restrict__ u,
                    const float* __restrict__ delta,
                    const float* __restrict__ A,
                    const float* __restrict__ Bm,
                    const float* __restrict__ Cm,
                    const float* __restrict__ Dv,
                    const float* __restrict__ z,
                    float* __restrict__ out)
{
  // double-buffered B/C panels shared by all 8 waves (same b, same chunk schedule)
  __shared__ __align__(16) float Bs[2][NN][CHUNK];   // 32 KB
  __shared__ __align__(16) float Cs[2][NN][CHUNK];   // 32 KB

  const int tid  = threadIdx.x;
  const int lane = tid & 31;
  const int wave = tid >> 5;
  const int b    = blockIdx.y;
  const int d    = blockIdx.x * WPB + wave;

  const float* Bbase = Bm + (size_t)b * NN * LL;
  const float* Cbase = Cm + (size_t)b * NN * LL;

  // Async global->LDS b128 panel fetch: 8 instructions per thread per chunk
  // -> ASYNCcnt += 8 per wave per issue().
  auto issue = [&](int chunk, int buf) {
    const int l0 = chunk * CHUNK;
#pragma unroll
    for (int k = 0; k < V4_PER_THREAD; ++k) {
      const int e4 = tid + k * THREADS;          // float4 slot in [0, 1024)
      const int n  = e4 / (CHUNK / 4);
      const int j4 = e4 % (CHUNK / 4);
      const float* gB = Bbase + (size_t)n * LL + l0 + j4 * 4;
      const float* gC = Cbase + (size_t)n * LL + l0 + j4 * 4;
      const uint32_t lB = (uint32_t)(uintptr_t)(&Bs[buf][n][j4 * 4]);
      const uint32_t lC = (uint32_t)(uintptr_t)(&Cs[buf][n][j4 * 4]);
      asm volatile("global_load_async_to_lds_b128 %0, %1, off"
                   :: "v"(lB), "v"(gB) : "memory");
      asm volatile("global_load_async_to_lds_b128 %0, %1, off"
                   :: "v"(lC), "v"(gC) : "memory");
    }
  };

  // per-d constants
  float Aa[NN];
#pragma unroll
  for (int n = 0; n < NN; ++n) Aa[n] = A[d * NN + n];
  const float Dval = Dv[d];

  // inter-chunk scan carry (wave-uniform, kept via v_readlane)
  float carry[NN];
#pragma unroll
  for (int n = 0; n < NN; ++n) carry[n] = 0.0f;

  const size_t row = ((size_t)b * DD + d) * LL;

  issue(0, 0);

  for (int c = 0; c < NCHUNK; ++c) {
    const int buf = c & 1;
    if (c + 1 < NCHUNK) {
      issue(c + 1, buf ^ 1);                             // prefetch next panel
      asm volatile("s_wait_asynccnt 0x8" ::: "memory");  // current panel landed
    } else {
      asm volatile("s_wait_asynccnt 0x0" ::: "memory");
    }
    __syncthreads();

    const int l0 = c * CHUNK;
    const float4* uP = reinterpret_cast<const float4*>(u     + row + l0);
    const float4* dP = reinterpret_cast<const float4*>(delta + row + l0);
    const float4* zP = reinterpret_cast<const float4*>(z     + row + l0);
    const float4 uva = uP[lane * 2], uvb = uP[lane * 2 + 1];
    const float4 dva = dP[lane * 2], dvb = dP[lane * 2 + 1];
    const float4 zva = zP[lane * 2], zvb = zP[lane * 2 + 1];

    const float uu[ILP] = {uva.x, uva.y, uva.z, uva.w, uvb.x, uvb.y, uvb.z, uvb.w};
    const float dd[ILP] = {dva.x, dva.y, dva.z, dva.w, dvb.x, dvb.y, dvb.z, dvb.w};
    const float zz[ILP] = {zva.x, zva.y, zva.z, zva.w, zvb.x, zvb.y, zvb.z, zvb.w};

    float dt[ILP], du[ILP], y[ILP];
#pragma unroll
    for (int i = 0; i < ILP; ++i) {
      dt[i] = softplus_f(dd[i]);
      du[i] = dt[i] * uu[i];
      y[i]  = 0.0f;
    }

#pragma unroll
    for (int n = 0; n < NN; ++n) {
      const float An = Aa[n];

      const float4* Brow = reinterpret_cast<const float4*>(&Bs[buf][n][0]);
      const float4 bA = Brow[lane * 2], bB = Brow[lane * 2 + 1];
      const float bl[ILP] = {bA.x, bA.y, bA.z, bA.w, bB.x, bB.y, bB.z, bB.w};

      float a[ILP], bb[ILP];
#pragma unroll
      for (int i = 0; i < ILP; ++i) {
        a[i]  = __expf(dt[i] * An);
        bb[i] = du[i] * bl[i];
      }

      // compose the lane's 8 recurrence ops into one (ca, cb)
      float ca = a[0], cb = bb[0];
#pragma unroll
      for (int i = 1; i < ILP; ++i) {
        cb = fmaf(a[i], cb, bb[i]);
        ca *= a[i];
      }

      // fold inter-chunk carry into lane 0's op: scan b-part then equals x_after
      if (lane == 0) cb = fmaf(ca, carry[n], cb);

      // wave32 inclusive Hillis-Steele scan; last step needs only the b-part
      float as = ca, bs = cb;
#pragma unroll
      for (int off = 1; off <= 8; off <<= 1) {
        const float ap = __shfl_up(as, off, 32);
        const float bp = __shfl_up(bs, off, 32);
        if (lane >= off) { bs = fmaf(as, bp, bs); as *= ap; }
      }
      {
        const float bp = __shfl_up(bs, 16, 32);
        if (lane >= 16) bs = fmaf(as, bp, bs);
      }
      // bs == state after this lane's last element (carry included)

      float state = __shfl_up(bs, 1, 32);   // state before this lane's elements
      if (lane == 0) state = carry[n];

      const float4* Crow = reinterpret_cast<const float4*>(&Cs[buf][n][0]);
      const float4 cA = Crow[lane * 2], cB = Crow[lane * 2 + 1];
      const float cl[ILP] = {cA.x, cA.y, cA.z, cA.w, cB.x, cB.y, cB.z, cB.w};

      float x = state;
#pragma unroll
      for (int i = 0; i < ILP - 1; ++i) {
        x = fmaf(a[i], x, bb[i]);
        y[i] = fmaf(x, cl[i], y[i]);
      }
      y[ILP - 1] = fmaf(bs, cl[ILP - 1], y[ILP - 1]);   // x7 == bs

      // broadcast lane 31's end-state as next chunk's carry (v_readlane, no DS op)
      carry[n] = __uint_as_float(
          __builtin_amdgcn_readlane(__float_as_uint(bs), 31));
    }

    float o[ILP];
#pragma unroll
    for (int i = 0; i < ILP; ++i) {
      o[i] = (y[i] + uu[i] * Dval) * (zz[i] * sigmoid_f(zz[i]));
    }
    float4* oP = reinterpret_cast<float4*>(out + row + l0);
    oP[lane * 2]     = make_float4(o[0], o[1], o[2], o[3]);
    oP[lane * 2 + 1] = make_float4(o[4], o[5], o[6], o[7]);

    __syncthreads();  // everyone done with buf before it is re-issued
  }
}

extern "C" void kernel_launch(void* const* d_in, const int* in_sizes, int n_in,
                              void* d_out, int out_size, void* d_ws, size_t ws_size,
                              hipStream_t stream) {
  // setup_inputs order: u, delta, A, B, C, D, z  (all float32)
  const float* u     = (const float*)d_in[0];
  const float* delta = (const float*)d_in[1];
  const float* A     = (const float*)d_in[2];
  const float* Bm    = (const float*)d_in[3];
  const float* Cm    = (const float*)d_in[4];
  const float* Dv    = (const float*)d_in[5];
  const float* z     = (const float*)d_in[6];
  float* out = (float*)d_out;
  (void)in_sizes; (void)n_in; (void)out_size; (void)d_ws; (void)ws_size;

  dim3 grid(DD / WPB, BB);   // (128, 2) blocks, 8 wave32 each -> 2048 (b,d) rows
  dim3 block(THREADS);
  hipLaunchKernelGGL(selscan_kernel, grid, block, 0, stream,
                     u, delta, A, Bm, Cm, Dv, z, out);
}